// AnchorPlusLoss_3822520894149
// MI455X (gfx1250) — compile-verified
//
#include <hip/hip_runtime.h>
#include <stdint.h>

typedef float v2f __attribute__((ext_vector_type(2)));
typedef float v8f __attribute__((ext_vector_type(8)));

#define BB 4
#define NN 1024
#define EE 32
#define TPB 128        // 4 waves of 32
#define WAVES 4
#define JT_PER_WAVE 16 // 64 j-tiles / 4 waves

// ---------------------------------------------------------------------------
// Kernel 0: augmented embedding W = emb (+ coords on dims 0,1) and row norms.
// One wave32 per row (E == 32 lanes exactly); wave shuffle reduction for ||W||^2.
// ---------------------------------------------------------------------------
__global__ void prep_kernel(const float* __restrict__ emb,
                            const float* __restrict__ coords,
                            float* __restrict__ W,
                            float* __restrict__ norm2) {
  int g   = blockIdx.x * blockDim.x + threadIdx.x;
  int row = g >> 5;     // global row index in [0, B*N)
  int e   = g & 31;     // element index, E == 32
  float v = emb[row * EE + e];
  if (e < 2) v += coords[row * 2 + e];
  W[row * EE + e] = v;
  float sq = v * v;
  #pragma unroll
  for (int off = 16; off > 0; off >>= 1)
    sq += __shfl_xor(sq, off, 32);
  if (e == 0) norm2[row] = sq;
}

// ---------------------------------------------------------------------------
// Kernel 1: one workgroup per (batch, 16-row i-tile).
// Phase 1 (only WMMA pass): for each of this wave's 16 j-tiles, compute the
//   16x16 Gram tile with 8x V_WMMA_F32_16X16X4_F32 (K=32), derive sim,
//   stash sim in LDS (64 KB of the WGP's 320 KB), pack mask bits into VGPRs,
//   and accumulate per-row sum(exp(sim)) over negatives.
// Phase 2: pure LDS replay — loss += log(exp(sim)+S_row) - sim on positives.
// ---------------------------------------------------------------------------
__global__ void anchor_loss_kernel(const float* __restrict__ W,
                                   const float* __restrict__ norm2,
                                   const uint8_t* __restrict__ mask,
                                   float* __restrict__ partial) {
  // sim cache: idx = t*8 + r in [0,128); [idx][tid] => lanes hit distinct banks
  __shared__ float simbuf[128 * TPB];          // 64 KB
  __shared__ float red[WAVES * 16 * 16];       // phase-1 rowsum, reused for loss
  __shared__ float sumexp_s[16];

  const int blk = blockIdx.x;        // 0 .. B*(N/16)-1 = 255
  const int b   = blk >> 6;          // / 64
  const int i0  = (blk & 63) * 16;

  const int tid  = threadIdx.x;
  const int wave = tid >> 5;
  const int lane = tid & 31;
  const int hi   = lane >> 4;        // A/B: K pair select; C/D: M 0..7 vs 8..15
  const int lo   = lane & 15;
  const int kb   = hi * 2;

  const float*   Wb = W + b * NN * EE;
  const float*   nb = norm2 + b * NN;
  const uint8_t* mb = mask + (size_t)b * NN * NN;

  // A operand (rows i0..i0+15), fixed for the whole block: 8 K-chunks of 4.
  v2f aCh[8];
  #pragma unroll
  for (int kc = 0; kc < 8; ++kc)
    aCh[kc] = *(const v2f*)(Wb + (i0 + lo) * EE + kc * 4 + kb);

  float ni[8];
  #pragma unroll
  for (int r = 0; r < 8; ++r) ni[r] = nb[i0 + r + 8 * hi];

  // ---- Phase 1: WMMA sweep; cache sim + mask bits; negative sumexp ----
  float accExp[8];
  #pragma unroll
  for (int r = 0; r < 8; ++r) accExp[r] = 0.0f;
  uint32_t mbits[4] = {0u, 0u, 0u, 0u};        // raw patch_mask bit per (t,r)

  for (int t = 0; t < JT_PER_WAVE; ++t) {
    const int j0 = (wave * JT_PER_WAVE + t) * 16;
    v8f g = {};
    #pragma unroll
    for (int kc = 0; kc < 8; ++kc) {
      v2f bv = *(const v2f*)(Wb + (j0 + lo) * EE + kc * 4 + kb);
      // (neg_a, A, neg_b, B, c_mod, C, reuse_a, reuse_b)
      g = __builtin_amdgcn_wmma_f32_16x16x4_f32(false, aCh[kc], false, bv,
                                                (short)0, g, false, false);
    }
    const int j = j0 + lo;
    const float njv = nb[j];
    #pragma unroll
    for (int r = 0; r < 8; ++r) {
      const int i = i0 + r + 8 * hi;
      float ssq = ni[r] + njv - 2.0f * g[r];
      ssq = ssq > 0.0f ? ssq : 0.0f;                 // clamp FP cancellation
      float dist = __builtin_sqrtf(ssq);
      float sim  = 1.0f / (1.0f + __expf(dist - 5.0f));  // sigmoid(5 - dist)

      simbuf[(t * 8 + r) * TPB + tid] = sim;          // conflict-free DS store

      uint32_t mraw = (mb[(size_t)i * NN + j] != 0) ? 1u : 0u;
      mbits[t >> 2] |= mraw << ((t * 8 + r) & 31);

      bool excl = (mraw != 0u) | (i == j);            // patch_mask | eye
      accExp[r] += excl ? 0.0f : __expf(sim);
    }
  }

  // reduce per-row negative sumexp across lanes/waves
  #pragma unroll
  for (int r = 0; r < 8; ++r)
    red[(wave * 16 + (r + 8 * hi)) * 16 + lo] = accExp[r];
  __syncthreads();
  if (tid < 16) {
    float s = 0.0f;
    for (int w = 0; w < WAVES; ++w)
      for (int l = 0; l < 16; ++l) s += red[(w * 16 + tid) * 16 + l];
    sumexp_s[tid] = s;   // S_row = sum over negatives of exp(sim)
  }
  __syncthreads();

  // ---- Phase 2: LDS replay, no global traffic, no WMMA recompute ----
  float Srow[8];
  #pragma unroll
  for (int r = 0; r < 8; ++r) Srow[r] = sumexp_s[r + 8 * hi];

  float lossLocal = 0.0f;
  for (int t = 0; t < JT_PER_WAVE; ++t) {
    #pragma unroll
    for (int r = 0; r < 8; ++r) {
      float sim = simbuf[(t * 8 + r) * TPB + tid];
      float c   = __logf(__expf(sim) + Srow[r]) - sim; // logaddexp - sim
      uint32_t bit = (mbits[t >> 2] >> ((t * 8 + r) & 31)) & 1u;
      lossLocal += bit ? c : 0.0f;
    }
  }

  // ---- Deterministic block reduction (reuse `red`) ----
  __syncthreads();           // red's phase-1 contents fully consumed
  red[tid] = lossLocal;
  __syncthreads();
  #pragma unroll
  for (int s = TPB / 2; s > 0; s >>= 1) {
    if (tid < s) red[tid] += red[tid + s];
    __syncthreads();
  }
  if (tid == 0) partial[blk] = red[0];
}

// ---------------------------------------------------------------------------
// Kernel 2: fixed-order reduction of the 256 per-block partials.
// ---------------------------------------------------------------------------
__global__ void final_reduce_kernel(const float* __restrict__ partial,
                                    float* __restrict__ out, int n) {
  __shared__ float s[256];
  int tid = threadIdx.x;
  s[tid] = (tid < n) ? partial[tid] : 0.0f;
  __syncthreads();
  #pragma unroll
  for (int k = 128; k > 0; k >>= 1) {
    if (tid < k) s[tid] += s[tid + k];
    __syncthreads();
  }
  if (tid == 0) out[0] = s[0];
}

// ---------------------------------------------------------------------------
// Launch: d_in[0]=embedding f32[4,1024,32], d_in[1]=abs_coords f32[4,1024,2],
// d_in[2]=patch_mask bool(u8)[4,1024,1024]; d_out = scalar f32 loss.
// Workspace: W (B*N*E f32) + norm2 (B*N f32) + partial (256 f32) ~ 530 KB.
// ---------------------------------------------------------------------------
extern "C" void kernel_launch(void* const* d_in, const int* in_sizes, int n_in,
                              void* d_out, int out_size, void* d_ws, size_t ws_size,
                              hipStream_t stream) {
  (void)in_sizes; (void)n_in; (void)out_size; (void)ws_size;
  const float*   emb    = (const float*)d_in[0];
  const float*   coords = (const float*)d_in[1];
  const uint8_t* maskp  = (const uint8_t*)d_in[2];
  float* out = (float*)d_out;

  float* ws      = (float*)d_ws;
  float* W       = ws;                       // B*N*E = 131072 floats
  float* norm2   = W + BB * NN * EE;         // B*N   =   4096 floats
  float* partial = norm2 + BB * NN;          // B*N/16 =   256 floats

  // Kernel 0: augmented embedding + row norms. One wave per row.
  prep_kernel<<<(BB * NN * EE) / 256, 256, 0, stream>>>(emb, coords, W, norm2);

  // Kernel 1: WMMA Gram tiles + cached-sim loss. 256 workgroups x 128 threads.
  anchor_loss_kernel<<<BB * (NN / 16), TPB, 0, stream>>>(W, norm2, maskp, partial);

  // Kernel 2: deterministic final sum.
  final_reduce_kernel<<<1, 256, 0, stream>>>(partial, out, BB * (NN / 16));
}